// VolumeRenderer_91242285236437
// MI455X (gfx1250) — compile-verified
//
#include <hip/hip_runtime.h>
#include <cstdint>

#ifndef __has_builtin
#define __has_builtin(x) 0
#endif

#if __has_builtin(__builtin_amdgcn_global_load_async_to_lds_b128)
#define HAVE_ASYNC_LDS 1
#else
#define HAVE_ASYNC_LDS 0
#endif

#define BIG_GAMMA 1e10f
#define EPS_T     1e-10f

// One wave32 per ray. Lane L owns samples [4L, 4L+4).
// Memory-bound: ~202 MB total @ 23.3 TB/s -> ~8.7us floor.
__global__ __launch_bounds__(256) void volrend_kernel(
    const float* __restrict__ rf,     // (R,128,4)
    const float* __restrict__ dirs,   // (R,3)
    const float* __restrict__ depth,  // (R,128)
    float* __restrict__ out,          // rgb (R*3) ++ weights (R*128)
    int R)
{
    const int lane = threadIdx.x & 31;
    const int wid  = threadIdx.x >> 5;
    const int ray  = blockIdx.x * 8 + wid;
    if (ray >= R) return;  // ray is wave-uniform; whole wave exits together

#if HAVE_ASYNC_LDS
    // Stage this ray's 2KB radiance tile into LDS with CDNA5 async DMA.
    // Store-side granule swizzle g -> ((g&3)<<5)|(g>>2) makes the read side a
    // plain ds_load_b128 at byte 16*lane + 512*k holding sample 4*lane+k,
    // with only the unavoidable 2-way bank conflict (512B per instruction).
    __shared__ float ldsbuf[8 * 512];  // 8 waves * 2KB
    float* wlds = ldsbuf + wid * 512;
    {
        // Builtin signature (revealed by clang diagnostic): b128 variant takes
        // pointers to 16-byte int vectors in AS1 (global) / AS3 (LDS).
        typedef int v4i __attribute__((ext_vector_type(4)));
        typedef __attribute__((address_space(1))) v4i* gptr_t;
        typedef __attribute__((address_space(3))) v4i* lptr_t;
        const float* base = rf + (size_t)ray * 512;
        #pragma unroll
        for (int it = 0; it < 4; ++it) {
            int g   = lane + 32 * it;              // linear 16B granule
            int swz = ((g & 3) << 5) | (g >> 2);   // swizzled granule
            __builtin_amdgcn_global_load_async_to_lds_b128(
                (gptr_t)(base + g * 4), (lptr_t)(wlds + swz * 4),
                /*offset=*/0, /*cpol=*/0);
        }
    }
#endif

    // --- depth diffs + ray-direction norm (overlaps async loads) ---
    const float* dRay = depth + (size_t)ray * 128;
    float4 d  = ((const float4*)dRay)[lane];       // fully coalesced 512B/wave
    float  dn = __shfl_down(d.x, 1, 32);           // next lane's first depth

    const float* dir = dirs + (size_t)ray * 3;
    float dx = dir[0], dy = dir[1], dz = dir[2];
    float dirn = sqrtf(dx * dx + dy * dy + dz * dz);

    float g0 = (d.y - d.x) * dirn;
    float g1 = (d.z - d.y) * dirn;
    float g2 = (d.w - d.z) * dirn;
    float g3 = ((lane == 31) ? BIG_GAMMA : (dn - d.w)) * dirn;

    // --- fetch radiance samples (r,g,b,sigma) for samples 4L..4L+3 ---
    float4 s0, s1, s2, s3;
#if HAVE_ASYNC_LDS
#if __has_builtin(__builtin_amdgcn_s_wait_asynccnt)
    __builtin_amdgcn_s_wait_asynccnt(0);
#else
    asm volatile("s_wait_asynccnt 0" ::: "memory");
#endif
    {
        const float4* lq = (const float4*)wlds;
        s0 = lq[lane + 0 * 32];
        s1 = lq[lane + 1 * 32];
        s2 = lq[lane + 2 * 32];
        s3 = lq[lane + 3 * 32];
    }
#else
    {
        const float4* rfc = (const float4*)(rf + (size_t)ray * 512) + lane * 4;
        s0 = rfc[0]; s1 = rfc[1]; s2 = rfc[2]; s3 = rfc[3];
    }
#endif

    // --- alpha / transmittance (match reference: t = 1 - alpha + EPS) ---
    float a0 = 1.0f - __expf(-s0.w * g0);
    float a1 = 1.0f - __expf(-s1.w * g1);
    float a2 = 1.0f - __expf(-s2.w * g2);
    float a3 = 1.0f - __expf(-s3.w * g3);

    float t0 = 1.0f - a0 + EPS_T;
    float t1 = 1.0f - a1 + EPS_T;
    float t2 = 1.0f - a2 + EPS_T;
    float t3 = 1.0f - a3 + EPS_T;

    // In-lane exclusive products over the 4-sample chunk.
    float e1  = t0;
    float e2  = t0 * t1;
    float e3  = e2 * t2;
    float tot = e3 * t3;

    // Wave-wide inclusive product scan of chunk totals (5 DS-permute steps).
    float p = tot;
    #pragma unroll
    for (int off = 1; off < 32; off <<= 1) {
        float v = __shfl_up(p, off, 32);
        if (lane >= off) p *= v;
    }
    float T0 = __shfl_up(p, 1, 32);   // exclusive prefix for this lane's chunk
    if (lane == 0) T0 = 1.0f;

    float w0 = T0 * a0;
    float w1 = (T0 * e1) * a1;
    float w2 = (T0 * e2) * a2;
    float w3 = (T0 * e3) * a3;

    // --- weighted RGB accumulation + wave reduction ---
    float cr = w0 * s0.x + w1 * s1.x + w2 * s2.x + w3 * s3.x;
    float cg = w0 * s0.y + w1 * s1.y + w2 * s2.y + w3 * s3.y;
    float cb = w0 * s0.z + w1 * s1.z + w2 * s2.z + w3 * s3.z;
    #pragma unroll
    for (int off = 16; off >= 1; off >>= 1) {
        cr += __shfl_xor(cr, off, 32);
        cg += __shfl_xor(cg, off, 32);
        cb += __shfl_xor(cb, off, 32);
    }

    // --- stores: weights are contiguous 16B per lane (coalesced b128) ---
    float* wout = out + (size_t)R * 3 + (size_t)ray * 128;
    ((float4*)wout)[lane] = make_float4(w0, w1, w2, w3);

    if (lane == 0) {
        float* ro = out + (size_t)ray * 3;
        ro[0] = cr; ro[1] = cg; ro[2] = cb;
    }
}

extern "C" void kernel_launch(void* const* d_in, const int* in_sizes, int n_in,
                              void* d_out, int out_size, void* d_ws, size_t ws_size,
                              hipStream_t stream) {
    const float* rf   = (const float*)d_in[0];  // (1,R,128,4) f32
    const float* dirs = (const float*)d_in[1];  // (1,R,3)     f32
    const float* dep  = (const float*)d_in[2];  // (1,R,128)   f32
    // d_in[3] = include_weights (==1); output buffer already sized for both.

    const int R = in_sizes[1] / 3;              // 65536
    const int blocks = (R + 7) / 8;             // 8 rays (waves) per 256-thr block
    volrend_kernel<<<blocks, 256, 0, stream>>>(rf, dirs, dep, (float*)d_out, R);
}